// RadiusGraph_15891378995592
// MI455X (gfx1250) — compile-verified
//
#include <hip/hip_runtime.h>
#include <hip/hip_bf16.h>

typedef __attribute__((ext_vector_type(2))) float v2f;
typedef __attribute__((ext_vector_type(4))) float v4f;
typedef __attribute__((ext_vector_type(8))) float v8f;

#define K_NB 32
#define R2   0.04f
#define FINF __builtin_inff()

#define TROWS 32            // destination rows per wave (two 16-row WMMA tiles)
#define TCOLS 32            // source cols per iteration (two 16-col sub-chunks)
#define PITCH 36            // LDS row pitch (floats): 16B-aligned, odd-bank spread

// ---------------------------------------------------------------------------
// Kernel 1: per-node prep.
//  sq[i] = |p_i|^2
//  Amat[i] = {x, y, z, 1}          (A operand rows, 16x4 layout friendly)
//  Bmat[i] = {-2x, -2y, -2z, sq}   (B operand cols: A.B = sq_j - 2 dot)
//  batch segment bounds via sorted-boundary detection (no atomics)
//  pass-through outputs: coords, batch
// ---------------------------------------------------------------------------
__global__ __launch_bounds__(256) void rg_prep_kernel(
    const float* __restrict__ coord, const int* __restrict__ batch,
    float* __restrict__ sq, float* __restrict__ Amat, float* __restrict__ Bmat,
    int* __restrict__ segLo, int* __restrict__ segHi,
    float* __restrict__ out_coord, int* __restrict__ out_batch, int N)
{
    int i = blockIdx.x * blockDim.x + threadIdx.x;
    if (i >= N) return;
    float x = coord[i * 3 + 0];
    float y = coord[i * 3 + 1];
    float z = coord[i * 3 + 2];
    float s = x * x + y * y + z * z;
    sq[i] = s;
    Amat[i * 4 + 0] = x;
    Amat[i * 4 + 1] = y;
    Amat[i * 4 + 2] = z;
    Amat[i * 4 + 3] = 1.0f;
    Bmat[i * 4 + 0] = -2.0f * x;
    Bmat[i * 4 + 1] = -2.0f * y;
    Bmat[i * 4 + 2] = -2.0f * z;
    Bmat[i * 4 + 3] = s;
    out_coord[i * 3 + 0] = x;
    out_coord[i * 3 + 1] = y;
    out_coord[i * 3 + 2] = z;
    int b = batch[i];
    out_batch[i] = b;
    // batch is sorted: record contiguous segment bounds
    if (i == 0 || batch[i - 1] != b) segLo[b] = i;
    if (i == N - 1 || batch[i + 1] != b) segHi[b] = i + 1;
}

// ---------------------------------------------------------------------------
// Kernel 2: radius + top-K via V_WMMA_F32_16X16X4_F32.
// One wave per 32 destination rows. Each iteration covers a 32x32 d2 tile
// with 4 WMMAs (2 row tiles x 2 col sub-chunks):
//   d2[m][n] = sq_m + sq_n - 2 dot(p_m, p_n)   (C operand = sq_m broadcast)
// WMMA epilogue applies only the radius clamp (flat cmp+cndmask) and stages
// the tile column-major in LDS with packed ds_store_b128 (v8f accumulator =
// 8 consecutive rows of one column). Each lane then owns one destination
// row; batch membership is an interval test on the sorted batch array
// (jcol in [segLo,segHi) of the row's batch), applied with the self test
// inside the rarely-taken "beats current worst" gate. Bubble insertion into
// a sorted register top-32 (strict <, ascending j => jax top_k tie order).
// ---------------------------------------------------------------------------
__global__ __launch_bounds__(32) void rg_topk_kernel(
    const float* __restrict__ Amat, const float* __restrict__ Bmat,
    const float* __restrict__ sq, const int* __restrict__ batch,
    const int* __restrict__ segLo, const int* __restrict__ segHi,
    int* __restrict__ edge_src, int* __restrict__ edge_dst,
    int* __restrict__ degree, int N)
{
    __shared__ __align__(16) float sh[TCOLS * PITCH];   // [col][row], 4.5 KB

    const int lane = threadIdx.x;        // 0..31
    const int half = lane >> 4;          // A/B operand half (K-split)
    const int lcol = lane & 15;          // column within a 16-wide sub-chunk
    const int i0 = blockIdx.x * TROWS;   // first destination row of this tile

    // A operands for the two stacked 16-row tiles
    const v2f* A2 = (const v2f*)Amat;
    v2f a0 = A2[(i0 + lcol) * 2 + half];
    v2f a1 = A2[(i0 + 16 + lcol) * 2 + half];

    // C operands (sq_m broadcast)
    v8f c0acc, c1acc;
    #pragma unroll
    for (int v = 0; v < 8; ++v) {
        int m = v + half * 8;
        c0acc[v] = sq[i0 + m];
        c1acc[v] = sq[i0 + 16 + m];
    }

    // This lane's destination row and its batch segment (interval test
    // replaces per-pair batch compares: batch is sorted & contiguous).
    const int i = i0 + lane;
    const int myB = batch[i];
    const int myLo = segLo[myB];
    const int myHi = segHi[myB];         // <= N, so clamped j>=N is rejected

    // Sorted (ascending d2) top-K state
    float best_d[K_NB];
    int   best_i[K_NB];
    #pragma unroll
    for (int k = 0; k < K_NB; ++k) { best_d[k] = FINF; best_i[k] = N; }

    // Contiguous source range covering every batch present in this row tile.
    const int lo = segLo[batch[i0]] & ~(TCOLS - 1);
    const int hi = segHi[batch[i0 + TROWS - 1]];

    const v2f* B2 = (const v2f*)Bmat;

    for (int cc0 = lo; cc0 < hi; cc0 += TCOLS) {
        #pragma unroll
        for (int sub = 0; sub < 2; ++sub) {
            const int t = sub * 16 + lcol;          // tile column 0..31
            const int j = cc0 + t;                  // global source index
            const int jl = j < N ? j : N - 1;       // clamped load index
            v2f b = B2[jl * 2 + half];
            __builtin_prefetch(&B2[(jl + 4 * TCOLS) * 2], 0, 0);

            v8f d0 = __builtin_amdgcn_wmma_f32_16x16x4_f32(
                false, a0, false, b, (short)0, c0acc, false, false);
            v8f d1 = __builtin_amdgcn_wmma_f32_16x16x4_f32(
                false, a1, false, b, (short)0, c1acc, false, false);

            // Radius clamp only: flat cmp+cndmask per element.
            #pragma unroll
            for (int v = 0; v < 8; ++v) {
                d0[v] = (d0[v] <= R2) ? d0[v] : FINF;
                d1[v] = (d1[v] <= R2) ? d1[v] : FINF;
            }
            // Column-major packed stores: 8 consecutive rows per b128 pair.
            v4f* p0 = (v4f*)&sh[t * PITCH + half * 8];
            p0[0] = __builtin_shufflevector(d0, d0, 0, 1, 2, 3);
            p0[1] = __builtin_shufflevector(d0, d0, 4, 5, 6, 7);
            v4f* p1 = (v4f*)&sh[t * PITCH + 16 + half * 8];
            p1[0] = __builtin_shufflevector(d1, d1, 0, 1, 2, 3);
            p1[1] = __builtin_shufflevector(d1, d1, 4, 5, 6, 7);
        }
        __syncthreads();

        // Selection: all 32 lanes active, lane == destination row it owns.
        for (int col = 0; col < TCOLS; ++col) {
            float dv = sh[col * PITCH + lane];
            if (dv < best_d[K_NB - 1]) {           // gate: beats current worst
                int jcol = cc0 + col;
                // batch interval + self test (bitwise &: flat cndmask chain)
                bool valid = (jcol >= myLo) & (jcol < myHi) & (jcol != i);
                if (valid) {
                    float cd = dv;
                    int ci = jcol;
                    #pragma unroll
                    for (int k = 0; k < K_NB; ++k) {
                        bool sw = cd < best_d[k];  // strict: stable tie order
                        float td = best_d[k]; int ti = best_i[k];
                        best_d[k] = sw ? cd : td;
                        best_i[k] = sw ? ci : ti;
                        cd = sw ? td : cd;
                        ci = sw ? ti : ci;
                    }
                }
            }
        }
        __syncthreads();        // LDS reused next iteration
    }

    int deg = 0;
    #pragma unroll
    for (int k = 0; k < K_NB; ++k) deg += (best_d[k] < FINF) ? 1 : 0;
    degree[i] = deg;
    #pragma unroll
    for (int k = 0; k < K_NB; ++k) {
        edge_src[i * K_NB + k] = (best_d[k] < FINF) ? best_i[k] : N;
        edge_dst[i * K_NB + k] = i;
    }
}

// ---------------------------------------------------------------------------
// Kernel 3: masked gather + scatter-mean. One block per destination node,
// one thread per feature channel (D=128). Features are L2-resident (8 MB).
// ---------------------------------------------------------------------------
__global__ __launch_bounds__(128) void rg_aggregate_kernel(
    const float* __restrict__ feat, const int* __restrict__ edge_src,
    const int* __restrict__ degree, float* __restrict__ out_feat,
    int N, int D)
{
    int i = blockIdx.x;
    int ch = threadIdx.x;
    if (ch >= D) return;
    int deg = degree[i];
    float acc = 0.0f;
    #pragma unroll 4
    for (int k = 0; k < K_NB; ++k) {
        int s = edge_src[i * K_NB + k];
        if (s < N) acc += feat[s * D + ch];
    }
    float dn = (float)(deg > 0 ? deg : 1);
    out_feat[i * D + ch] = acc / dn;
}

// ---------------------------------------------------------------------------
extern "C" void kernel_launch(void* const* d_in, const int* in_sizes, int n_in,
                              void* d_out, int out_size, void* d_ws, size_t ws_size,
                              hipStream_t stream) {
    const float* coord = (const float*)d_in[0];   // [N,3]
    const float* feat  = (const float*)d_in[1];   // [N,D]
    const int*   batch = (const int*)d_in[2];     // [N]

    const int N = in_sizes[0] / 3;
    const int D = in_sizes[1] / N;

    // Workspace layout (bytes)
    char* ws = (char*)d_ws;
    float* sq    = (float*)ws;                       ws += (size_t)N * 4;
    float* Amat  = (float*)ws;                       ws += (size_t)N * 16;
    float* Bmat  = (float*)ws;                       ws += (size_t)N * 16;
    int*   segLo = (int*)ws;                         ws += 64 * 4;
    int*   segHi = (int*)ws;                         ws += 64 * 4;

    // Output layout (flat, return order): coords | feat | esrc | edst | deg | batch
    char* ob = (char*)d_out;
    float* out_coord = (float*)ob;                              ob += (size_t)N * 3 * 4;
    float* out_feat  = (float*)ob;                              ob += (size_t)N * D * 4;
    int*   out_esrc  = (int*)ob;                                ob += (size_t)N * K_NB * 4;
    int*   out_edst  = (int*)ob;                                ob += (size_t)N * K_NB * 4;
    int*   out_deg   = (int*)ob;                                ob += (size_t)N * 4;
    int*   out_batch = (int*)ob;

    rg_prep_kernel<<<(N + 255) / 256, 256, 0, stream>>>(
        coord, batch, sq, Amat, Bmat, segLo, segHi, out_coord, out_batch, N);

    rg_topk_kernel<<<N / TROWS, 32, 0, stream>>>(
        Amat, Bmat, sq, batch, segLo, segHi, out_esrc, out_edst, out_deg, N);

    rg_aggregate_kernel<<<N, 128, 0, stream>>>(
        feat, out_esrc, out_deg, out_feat, N, D);
}